// KernelConv_21689584845340
// MI455X (gfx1250) — compile-verified
//
#include <hip/hip_runtime.h>

typedef float v2f __attribute__((ext_vector_type(2)));
typedef float v8f __attribute__((ext_vector_type(8)));

#define EPSF 1e-8f
#define KDIM 164      // 3 len + 9 angle + 96 support + 24 edge + 32 center
#define KSTEPS 41     // KDIM / 4
#define NTPL 256      // 32 l * 8 perm-slots (6 real perms + 2 duplicates)
#define ROWS 128      // samples per block (8 waves * 16-col tiles)
#define LOUT 32

__device__ __constant__ int PERM3[6][3] = {
    {0,1,2},{0,2,1},{1,0,2},{1,2,0},{2,0,1},{2,1,0}
};

__global__ void __launch_bounds__(256) perm_match_kernel(
    const float* __restrict__ x_focal,      // [N][32]
    const float* __restrict__ p_focal,      // [N][3]
    const float* __restrict__ x_neighbor,   // [N][3][32]
    const float* __restrict__ p_neighbor,   // [N][3][3]
    const float* __restrict__ edge_nei,     // [N][3][8]
    const float* __restrict__ x_center,     // [32][32]
    const float* __restrict__ x_support,    // [32][3][32]
    const float* __restrict__ edge_sup,     // [32][3][8]
    const float* __restrict__ p_support,    // [32][3][3]
    const float* __restrict__ w_len,
    const float* __restrict__ w_ang,
    const float* __restrict__ w_cen,
    const float* __restrict__ w_sup,
    const float* __restrict__ w_edg,
    float* __restrict__ out,                // [N][32]
    int n_total)
{
    extern __shared__ float smem[];
    float* g_lds = smem;               // [NTPL][KDIM]  weighted templates
    float* f_lds = smem + NTPL*KDIM;   // [ROWS][KDIM]  normalized samples
    const int tid = threadIdx.x;

    // ---------------- Phase A: build 256 weighted template rows ----------
    {
        const int t = tid;             // 0..255, one template per thread
        const int l = t >> 3;
        const int p = (t & 7) % 6;     // slots 6,7 duplicate perms 0,1 (max-safe)
        float* g = g_lds + t * KDIM;
        const float lw = w_len[0];
        const float aw = w_ang[0] * (1.0f/3.0f);
        const float sw = w_sup[0] * (1.0f/3.0f);
        const float ew = w_edg[0] * (1.0f/3.0f);
        const float cw = w_cen[0];

        float ls[3], psv[3][3];
        for (int d = 0; d < 3; ++d) {
            const int sd = PERM3[p][d];
            const float* pp = p_support + (l*3 + sd)*3;
            float a = pp[0], b = pp[1], c = pp[2];
            psv[d][0] = a; psv[d][1] = b; psv[d][2] = c;
            ls[d] = sqrtf(a*a + b*b + c*c);
        }
        // length part: length_w * nrm(len_sup)
        {
            float inv = lw / fmaxf(sqrtf(ls[0]*ls[0]+ls[1]*ls[1]+ls[2]*ls[2]), EPSF);
            g[0] = ls[0]*inv; g[1] = ls[1]*inv; g[2] = ls[2]*inv;
        }
        // angle part: (angle_w/3) * ps_hat
        for (int d = 0; d < 3; ++d) {
            float inv = aw / fmaxf(ls[d], EPSF);
            g[3 + d*3 + 0] = psv[d][0]*inv;
            g[3 + d*3 + 1] = psv[d][1]*inv;
            g[3 + d*3 + 2] = psv[d][2]*inv;
        }
        // support part: (support_w/3) * nrm(x_support row)
        for (int d = 0; d < 3; ++d) {
            const int sd = PERM3[p][d];
            const float* xr = x_support + (l*3 + sd)*32;
            float s = 0.0f;
            for (int a = 0; a < 32; ++a) s += xr[a]*xr[a];
            float inv = sw / fmaxf(sqrtf(s), EPSF);
            for (int a = 0; a < 32; ++a) g[12 + d*32 + a] = xr[a]*inv;
        }
        // edge part: (edge_w/3) * nrm(edge_support row)
        for (int d = 0; d < 3; ++d) {
            const int sd = PERM3[p][d];
            const float* er = edge_sup + (l*3 + sd)*8;
            float s = 0.0f;
            for (int e = 0; e < 8; ++e) s += er[e]*er[e];
            float inv = ew / fmaxf(sqrtf(s), EPSF);
            for (int e = 0; e < 8; ++e) g[108 + d*8 + e] = er[e]*inv;
        }
        // center part (perm-invariant, folded pre-max): center_w * nrm(x_center[l])
        {
            const float* xc = x_center + l*32;
            float s = 0.0f;
            for (int a = 0; a < 32; ++a) s += xc[a]*xc[a];
            float inv = cw / fmaxf(sqrtf(s), EPSF);
            for (int a = 0; a < 32; ++a) g[132 + a] = xc[a]*inv;
        }
    }

    // ---------------- Phase B: build 128 normalized sample rows ----------
    if (tid < ROWS) {
        const int n = blockIdx.x * ROWS + tid;
        float* f = f_lds + tid * KDIM;
        if (n < n_total) {
            const float* pf = p_focal + n*3;
            float p0 = pf[0], p1 = pf[1], p2 = pf[2];
            float len[3], pn[3][3];
            for (int d = 0; d < 3; ++d) {
                const float* q = p_neighbor + (n*3 + d)*3;
                float a = q[0]-p0, b = q[1]-p1, c = q[2]-p2;
                pn[d][0] = a; pn[d][1] = b; pn[d][2] = c;
                len[d] = sqrtf(a*a + b*b + c*c);
            }
            {
                float inv = 1.0f / fmaxf(sqrtf(len[0]*len[0]+len[1]*len[1]+len[2]*len[2]), EPSF);
                f[0] = len[0]*inv; f[1] = len[1]*inv; f[2] = len[2]*inv;
            }
            for (int d = 0; d < 3; ++d) {
                float inv = 1.0f / fmaxf(len[d], EPSF);
                f[3 + d*3 + 0] = pn[d][0]*inv;
                f[3 + d*3 + 1] = pn[d][1]*inv;
                f[3 + d*3 + 2] = pn[d][2]*inv;
            }
            for (int d = 0; d < 3; ++d) {
                const float* xr = x_neighbor + (n*3 + d)*32;
                float s = 0.0f;
                for (int a = 0; a < 32; ++a) s += xr[a]*xr[a];
                float inv = 1.0f / fmaxf(sqrtf(s), EPSF);
                for (int a = 0; a < 32; ++a) f[12 + d*32 + a] = xr[a]*inv;
            }
            for (int d = 0; d < 3; ++d) {
                const float* er = edge_nei + (n*3 + d)*8;
                float s = 0.0f;
                for (int e = 0; e < 8; ++e) s += er[e]*er[e];
                float inv = 1.0f / fmaxf(sqrtf(s), EPSF);
                for (int e = 0; e < 8; ++e) f[108 + d*8 + e] = er[e]*inv;
            }
            {
                const float* xf = x_focal + n*32;
                float s = 0.0f;
                for (int a = 0; a < 32; ++a) s += xf[a]*xf[a];
                float inv = 1.0f / fmaxf(sqrtf(s), EPSF);
                for (int a = 0; a < 32; ++a) f[132 + a] = xf[a]*inv;
            }
        } else {
            for (int k = 0; k < KDIM; ++k) f[k] = 0.0f;   // dead rows are benign
        }
    }
    __syncthreads();

    // ---------------- Phase C: WMMA GEMM, templates on M, samples on N ----
    // C = G_tile(16 templates x K) * F_tile(K x 16 samples):
    //   lane holds column n = j and rows M = v + 8*half  (v = acc index)
    //   => the 8 perm-slots of l-group (2*tt + half) live in this lane's 8
    //      accumulators; perm-max is 7 v_max_f32, no cross-lane traffic.
    const int lane = tid & 31;
    const int wv   = tid >> 5;         // wave id -> which 16-sample tile of F
    const int j    = lane & 15;        // M index (A) / N index (B)
    const int half = lane >> 4;        // K-pair select (k0 + 2*half, +1)

    // Preload the 41 B-fragments (samples) -- invariant across template tiles.
    const float* fbase = f_lds + (wv*16 + j)*KDIM + 2*half;
    v2f fb[KSTEPS];
    #pragma unroll
    for (int s = 0; s < KSTEPS; ++s) fb[s] = *(const v2f*)(fbase + 4*s);

    const int n_out = blockIdx.x * ROWS + wv*16 + j;   // sample this lane owns

    for (int tt = 0; tt < 16; ++tt) {  // 16 tiles x 16 templates = 256
        const float* gbase = g_lds + (tt*16 + j)*KDIM + 2*half;

        // Stage the whole A tile (template fragments) in registers so the
        // WMMA chain runs back-to-back with a single dscnt wait, and next
        // tile's loads can overlap the drain of this chain.
        v2f ga[KSTEPS];
        #pragma unroll
        for (int s = 0; s < KSTEPS; ++s) ga[s] = *(const v2f*)(gbase + 4*s);

        v8f acc0 = {0.f,0.f,0.f,0.f,0.f,0.f,0.f,0.f};
        v8f acc1 = {0.f,0.f,0.f,0.f,0.f,0.f,0.f,0.f};
        #pragma unroll
        for (int s = 0; s < KSTEPS; s += 2) {
            acc0 = __builtin_amdgcn_wmma_f32_16x16x4_f32(
                false, ga[s], false, fb[s], (short)0, acc0, false, false);
            if (s + 1 < KSTEPS) {
                acc1 = __builtin_amdgcn_wmma_f32_16x16x4_f32(
                    false, ga[s+1], false, fb[s+1], (short)0, acc1, false, false);
            }
        }
        v8f acc = acc0 + acc1;
        // perm-max over the 8 slots of this lane's l-group
        float m01 = fmaxf(acc[0], acc[1]);
        float m23 = fmaxf(acc[2], acc[3]);
        float m45 = fmaxf(acc[4], acc[5]);
        float m67 = fmaxf(acc[6], acc[7]);
        float m = fmaxf(fmaxf(m01, m23), fmaxf(m45, m67));
        if (n_out < n_total) out[n_out*LOUT + tt*2 + half] = m;
    }
}

extern "C" void kernel_launch(void* const* d_in, const int* in_sizes, int n_in,
                              void* d_out, int out_size, void* d_ws, size_t ws_size,
                              hipStream_t stream) {
    (void)n_in; (void)out_size; (void)d_ws; (void)ws_size;
    const float* x_focal    = (const float*)d_in[0];
    const float* p_focal    = (const float*)d_in[1];
    const float* x_neighbor = (const float*)d_in[2];
    const float* p_neighbor = (const float*)d_in[3];
    const float* edge_nei   = (const float*)d_in[4];
    const float* x_center   = (const float*)d_in[5];
    const float* x_support  = (const float*)d_in[6];
    const float* edge_sup   = (const float*)d_in[7];
    const float* p_support  = (const float*)d_in[8];
    const float* w_len      = (const float*)d_in[9];
    const float* w_ang      = (const float*)d_in[10];
    const float* w_cen      = (const float*)d_in[11];
    const float* w_sup      = (const float*)d_in[12];
    const float* w_edg      = (const float*)d_in[13];
    float* out = (float*)d_out;

    const int N = in_sizes[0] / 32;              // x_focal is [N][32]
    const int blocks = (N + ROWS - 1) / ROWS;
    const size_t smem_bytes = (size_t)(NTPL + ROWS) * KDIM * sizeof(float);

    perm_match_kernel<<<blocks, 256, smem_bytes, stream>>>(
        x_focal, p_focal, x_neighbor, p_neighbor, edge_nei,
        x_center, x_support, edge_sup, p_support,
        w_len, w_ang, w_cen, w_sup, w_edg,
        out, N);
}